// GNN_3358664426320
// MI455X (gfx1250) — compile-verified
//
#include <hip/hip_runtime.h>
#include <hip/hip_bf16.h>

#define N_NODES   100000
#define N_EDGES   3200000
#define D_FEAT    128
#define HIDDEN    16
#define N_CLASSES 8

typedef __attribute__((ext_vector_type(2))) float v2f;
typedef __attribute__((ext_vector_type(8))) float v8f;

// ---------------------------------------------------------------------------
// Init: cnt[i] = 1 (self-loop), zero the layer-1 accumulator and d_out.
// One grid covering the largest array (N_NODES*HIDDEN).
// ---------------------------------------------------------------------------
__global__ void k_init(unsigned* __restrict__ cnt,
                       float* __restrict__ a1,
                       float* __restrict__ out) {
    int i = blockIdx.x * blockDim.x + threadIdx.x;
    if (i < N_NODES) cnt[i] = 1u;
    if (i < N_NODES * HIDDEN) a1[i] = 0.0f;
    if (i < N_NODES * N_CLASSES) out[i] = 0.0f;
}

// ---------------------------------------------------------------------------
// Degree count over dst (self-loops already seeded as 1).
// ---------------------------------------------------------------------------
__global__ void k_count(const int* __restrict__ dst, unsigned* __restrict__ cnt) {
    int e = blockIdx.x * blockDim.x + threadIdx.x;
    if (e < N_EDGES) {
        int d = dst[e];
        atomicAdd(&cnt[d], 1u);
    }
}

__global__ void k_dis(const unsigned* __restrict__ cnt, float* __restrict__ dis) {
    int i = blockIdx.x * blockDim.x + threadIdx.x;
    if (i < N_NODES) dis[i] = rsqrtf((float)cnt[i]);
}

// ---------------------------------------------------------------------------
// Layer-1 GEMM: h1[100000,16] = x[100000,128] @ W1[128,16] via
// V_WMMA_F32_16X16X4_F32. One wave32 per 16-row tile, K accumulated in
// steps of 4 (32 WMMA ops/tile). W1 staged in LDS (8 KB).
//
// A frag  (16x4):  lane L, vgpr v  <-> A[L%16][kb + 2*(L>>4) + v]
// B frag  (4x16):  lane L, vgpr v  <-> B[kb + 2*(L>>4) + v][L%16]
// D frag (16x16):  lane L, vgpr r  <-> D[r + 8*(L>>4)][L%16]
// ---------------------------------------------------------------------------
__global__ void k_gemm1_wmma(const float* __restrict__ x,
                             const float* __restrict__ W1,
                             float* __restrict__ h1) {
    __shared__ float w_lds[D_FEAT * HIDDEN];   // 8 KB
    for (int i = threadIdx.x; i < D_FEAT * HIDDEN; i += blockDim.x)
        w_lds[i] = W1[i];
    __syncthreads();

    const int lane = threadIdx.x & 31;
    const int wave = threadIdx.x >> 5;
    const int tile = blockIdx.x * (blockDim.x >> 5) + wave;   // 16-row tile id
    if (tile >= N_NODES / 16) return;                         // wave-uniform exit

    const int row  = lane & 15;      // M (for A/D) or N (for B)
    const int half = lane >> 4;      // 0|1 -> K sub-pair selector
    const int m0   = tile * 16;

    const float* __restrict__ xrow = x + (size_t)(m0 + row) * D_FEAT + 2 * half;

    v8f acc = {};
    #pragma unroll 4
    for (int kb = 0; kb < D_FEAT; kb += 4) {
        // A: two contiguous f32 -> single b64 load
        v2f a = *(const v2f*)(xrow + kb);
        // B: from LDS, rows kB and kB+1 of W1, column `row`
        const int kB = kb + 2 * half;
        v2f b;
        b.x = w_lds[(kB + 0) * HIDDEN + row];
        b.y = w_lds[(kB + 1) * HIDDEN + row];
        acc = __builtin_amdgcn_wmma_f32_16x16x4_f32(
                  /*neg_a=*/false, a, /*neg_b=*/false, b,
                  /*c_mod=*/(short)0, acc,
                  /*reuse_a=*/false, /*reuse_b=*/false);
    }

    float* __restrict__ hbase = h1 + (size_t)(m0 + 8 * half) * HIDDEN + row;
    #pragma unroll
    for (int r = 0; r < 8; ++r)
        hbase[(size_t)r * HIDDEN] = acc[r];
}

// ---------------------------------------------------------------------------
// Edge aggregation: out[dst] += h[src] * dis[src]*dis[dst].
// F lanes per edge (F = 16 or 8); last N_NODES "edges" are the self-loops.
// ---------------------------------------------------------------------------
template <int F>
__global__ void k_aggregate(const float* __restrict__ h,
                            const int* __restrict__ src,
                            const int* __restrict__ dst,
                            const float* __restrict__ dis,
                            float* __restrict__ out) {
    const long long t = (long long)blockIdx.x * blockDim.x + threadIdx.x;
    const long long e = t / F;
    const int f = (int)(t - e * F);
    if (e >= (long long)N_EDGES + N_NODES) return;

    int s, d;
    if (e < N_EDGES) { s = src[e]; d = dst[e]; }
    else             { s = d = (int)(e - N_EDGES); }   // self-loop

    const float norm = dis[s] * dis[d];
    const float v = h[(size_t)s * F + f] * norm;
    unsafeAtomicAdd(&out[(size_t)d * F + f], v);       // global_atomic_add_f32
}

// ---------------------------------------------------------------------------
// Bias + ReLU for layer 1 (in place on a1).
// ---------------------------------------------------------------------------
__global__ void k_bias_relu(float* __restrict__ a1, const float* __restrict__ b1) {
    int i = blockIdx.x * blockDim.x + threadIdx.x;
    if (i < N_NODES * HIDDEN) {
        float v = a1[i] + b1[i & (HIDDEN - 1)];
        a1[i] = v > 0.0f ? v : 0.0f;
    }
}

// ---------------------------------------------------------------------------
// Layer-2 GEMM: h2[100000,8] = a1[100000,16] @ W2[16,8]  (tiny; plain VALU).
// ---------------------------------------------------------------------------
__global__ void k_gemm2(const float* __restrict__ a1,
                        const float* __restrict__ W2,
                        float* __restrict__ h2) {
    int i = blockIdx.x * blockDim.x + threadIdx.x;
    if (i >= N_NODES * N_CLASSES) return;
    const int n = i >> 3;
    const int c = i & 7;
    const float* __restrict__ row = a1 + (size_t)n * HIDDEN;
    float s = 0.0f;
    #pragma unroll
    for (int k = 0; k < HIDDEN; ++k)
        s = fmaf(row[k], W2[k * N_CLASSES + c], s);
    h2[i] = s;
}

// ---------------------------------------------------------------------------
// Final: logits = agg2 + b2 ; write log_softmax in place (d_out).
// ---------------------------------------------------------------------------
__global__ void k_log_softmax(float* __restrict__ out, const float* __restrict__ b2) {
    int n = blockIdx.x * blockDim.x + threadIdx.x;
    if (n >= N_NODES) return;
    float v[N_CLASSES];
    float mx = -3.402823466e+38f;
    #pragma unroll
    for (int c = 0; c < N_CLASSES; ++c) {
        v[c] = out[(size_t)n * N_CLASSES + c] + b2[c];
        mx = fmaxf(mx, v[c]);
    }
    float se = 0.0f;
    #pragma unroll
    for (int c = 0; c < N_CLASSES; ++c) se += expf(v[c] - mx);
    const float lse = mx + logf(se);
    #pragma unroll
    for (int c = 0; c < N_CLASSES; ++c)
        out[(size_t)n * N_CLASSES + c] = v[c] - lse;
}

// ---------------------------------------------------------------------------
// Host launcher
// ---------------------------------------------------------------------------
extern "C" void kernel_launch(void* const* d_in, const int* in_sizes, int n_in,
                              void* d_out, int out_size, void* d_ws, size_t ws_size,
                              hipStream_t stream) {
    const float* x  = (const float*)d_in[0];
    const int*   ei = (const int*)d_in[1];          // [2, E] int32
    const float* W1 = (const float*)d_in[2];
    const float* b1 = (const float*)d_in[3];
    const float* W2 = (const float*)d_in[4];
    const float* b2 = (const float*)d_in[5];
    float* out = (float*)d_out;

    const int* src = ei;
    const int* dst = ei + N_EDGES;

    char* ws = (char*)d_ws;
    unsigned* cnt = (unsigned*)(ws);                       //  0.4 MB
    float*    dis = (float*)(ws + (1u << 20));             //  0.4 MB
    float*    h1  = (float*)(ws + (2u << 20));             //  6.4 MB
    float*    a1  = (float*)(ws + (10u << 20));            //  6.4 MB
    float*    h2  = (float*)(ws + (18u << 20));            //  3.2 MB

    const int TPB = 256;

    // 0) init cnt=1, a1=0, d_out=0
    {
        int total = N_NODES * HIDDEN;
        k_init<<<(total + TPB - 1) / TPB, TPB, 0, stream>>>(cnt, a1, out);
    }
    // 1) degree count + inverse-sqrt
    k_count<<<(N_EDGES + TPB - 1) / TPB, TPB, 0, stream>>>(dst, cnt);
    k_dis<<<(N_NODES + TPB - 1) / TPB, TPB, 0, stream>>>(cnt, dis);

    // 2) h1 = x @ W1 via WMMA (one wave per 16-row tile, 8 waves/block)
    {
        const int tiles = N_NODES / 16;                    // 6250
        const int waves_per_block = TPB / 32;              // 8
        const int blocks = (tiles + waves_per_block - 1) / waves_per_block;
        k_gemm1_wmma<<<blocks, TPB, 0, stream>>>(x, W1, h1);
    }

    // 3) aggregate layer 1: a1[dst] += h1[src] * norm   (16 lanes / edge)
    {
        const long long total = (long long)(N_EDGES + N_NODES) * HIDDEN;
        const long long blocks = (total + TPB - 1) / TPB;
        k_aggregate<HIDDEN><<<(int)blocks, TPB, 0, stream>>>(h1, src, dst, dis, a1);
    }

    // 4) a1 = relu(a1 + b1)
    k_bias_relu<<<(N_NODES * HIDDEN + TPB - 1) / TPB, TPB, 0, stream>>>(a1, b1);

    // 5) h2 = a1 @ W2
    k_gemm2<<<(N_NODES * N_CLASSES + TPB - 1) / TPB, TPB, 0, stream>>>(a1, W2, h2);

    // 6) aggregate layer 2 into d_out (8 lanes / edge)
    {
        const long long total = (long long)(N_EDGES + N_NODES) * N_CLASSES;
        const long long blocks = (total + TPB - 1) / TPB;
        k_aggregate<N_CLASSES><<<(int)blocks, TPB, 0, stream>>>(h2, src, dst, dis, out);
    }

    // 7) + b2, log_softmax in place
    k_log_softmax<<<(N_NODES + TPB - 1) / TPB, TPB, 0, stream>>>(out, b2);
}